// ArcMarginSoftmaxWithLoss_12532714570064
// MI455X (gfx1250) — compile-verified
//
#include <hip/hip_runtime.h>
#include <stdint.h>

#ifndef __has_builtin
#define __has_builtin(x) 0
#endif

// ---------------- tunables ----------------
#define CHUNKS      8            // chunks per row -> 4096 workgroups
#define BLOCK       256          // 8 waves of 32
#define TILE_FLOATS 2048         // 8 KB tile, 8 floats per thread
#define HALF_TILE   (TILE_FLOATS / 2)
#define NBUF        4            // 4-deep LDS pipeline (32 KB)
#define DEPTH       3            // issue-ahead tiles; must be <= NBUF-1

// ---------------- reference constants (base-2 folded) ----------------
#define ARC_S      30.0f
#define K_S_LOG2E  43.28085122666891f    // S * log2(e)
#define LN2F       0.6931471805599453f
#define ARC_COS_M  0.8775825618903728f   // cos(0.5)
#define ARC_SIN_M  0.479425538604203f    // sin(0.5)
#define ARC_TH    (-0.8775825618903728f) // cos(pi - 0.5)
#define ARC_MM     0.23971276930210156f  // sin(pi - 0.5) * 0.5

// Probed builtin signature: (v4i addrspace(1)*, v4i addrspace(3)*, imm, imm)
typedef int v4i __attribute__((ext_vector_type(4)));
typedef __attribute__((address_space(1))) v4i as1_v4i;
typedef __attribute__((address_space(3))) v4i as3_v4i;

// Async global -> LDS copy of 16 bytes per lane (CDNA5 ASYNCcnt path).
__device__ __forceinline__ void async_b128(const float* g, float* l) {
#if __has_builtin(__builtin_amdgcn_global_load_async_to_lds_b128)
  __builtin_amdgcn_global_load_async_to_lds_b128(
      (as1_v4i*)(uintptr_t)g,
      (as3_v4i*)(uint32_t)(uintptr_t)l,   // low 32 bits of flat LDS addr == LDS offset
      0, 0);
#else
  asm volatile("global_load_async_to_lds_b128 %0, %1, off"
               :: "v"((uint32_t)(uintptr_t)l), "v"(g)
               : "memory");
#endif
}

template <int N>
__device__ __forceinline__ void wait_asynccnt() {
#if __has_builtin(__builtin_amdgcn_s_wait_asynccnt)
  __builtin_amdgcn_s_wait_asynccnt(N);
#else
  asm volatile("s_wait_asynccnt %0" :: "n"(N) : "memory");
#endif
  asm volatile("" ::: "memory");   // keep LDS reads behind the wait
}

// (m,s) <- combine with (om, os):  s = s*2^(m-mn) + os*2^(om-mn)
__device__ __forceinline__ void lse_combine(float& m, float& s, float om, float os) {
  float mn = fmaxf(m, om);
  s = s * exp2f(m - mn) + os * exp2f(om - mn);
  m = mn;
}

// ---------------------------------------------------------------------------
// Phase 1: per (row, chunk) online logsumexp of y = K * cos_theta
// (target column NOT special-cased here; fixed up in phase 2)
// ---------------------------------------------------------------------------
__global__ __launch_bounds__(BLOCK)
void arc_lse_partial_kernel(const float* __restrict__ cosT,
                            float2* __restrict__ partial,
                            int C, int chunkLen)
{
  __shared__ __align__(16) float lds[NBUF][TILE_FLOATS];
  __shared__ float red_m[BLOCK / 32];
  __shared__ float red_s[BLOCK / 32];

  const int row   = blockIdx.y;
  const int chunk = blockIdx.x;
  const int tid   = threadIdx.x;

  const int start = chunk * chunkLen;
  const int end   = min(C, start + chunkLen);
  const int n     = (end > start) ? (end - start) : 0;
  const float* src = cosT + (size_t)row * (size_t)C + start;

  float m = -3.0e38f;   // finite sentinel avoids (-inf) - (-inf) NaNs
  float s = 0.0f;

  // async b128 needs 16B-aligned addresses; true when C and chunkLen are %4==0
  const bool aligned = (((C & 3) == 0) && ((chunkLen & 3) == 0));
  const int T = aligned ? (n / TILE_FLOATS) : 0;

  // each wave copies its own 2x512B slice of a tile -> no cross-wave barriers
  auto issue_tile = [&](int t) {
    const int b = t & (NBUF - 1);
    const float* g = src + (size_t)t * TILE_FLOATS;
    async_b128(g + tid * 4,             &lds[b][tid * 4]);
    async_b128(g + HALF_TILE + tid * 4, &lds[b][HALF_TILE + tid * 4]);
  };

  if (T > 0) {
    // prologue: up to DEPTH tiles in flight
    const int pre = (T < DEPTH) ? T : DEPTH;
    for (int p = 0; p < pre; ++p) issue_tile(p);

    for (int t = 0; t < T; ++t) {
      if (t + DEPTH < T) issue_tile(t + DEPTH);

      // tiles issued beyond t: ahead = min(T-1-t, DEPTH); each tile = 2 ops.
      // waiting ASYNCcnt <= 2*ahead guarantees tile t is complete (in-order).
      const int ahead = min(T - 1 - t, DEPTH);
      if      (ahead >= 3) wait_asynccnt<6>();
      else if (ahead == 2) wait_asynccnt<4>();
      else if (ahead == 1) wait_asynccnt<2>();
      else                 wait_asynccnt<0>();

      const int buf = t & (NBUF - 1);
      float4 a = *(const float4*)&lds[buf][tid * 4];
      float4 b = *(const float4*)&lds[buf][HALF_TILE + tid * 4];

      float y0 = a.x * K_S_LOG2E, y1 = a.y * K_S_LOG2E;
      float y2 = a.z * K_S_LOG2E, y3 = a.w * K_S_LOG2E;
      float y4 = b.x * K_S_LOG2E, y5 = b.y * K_S_LOG2E;
      float y6 = b.z * K_S_LOG2E, y7 = b.w * K_S_LOG2E;

      float g8 = fmaxf(fmaxf(fmaxf(y0, y1), fmaxf(y2, y3)),
                       fmaxf(fmaxf(y4, y5), fmaxf(y6, y7)));
      float mn = fmaxf(m, g8);
      s = s * exp2f(m - mn)
        + ((exp2f(y0 - mn) + exp2f(y1 - mn)) + (exp2f(y2 - mn) + exp2f(y3 - mn)))
        + ((exp2f(y4 - mn) + exp2f(y5 - mn)) + (exp2f(y6 - mn) + exp2f(y7 - mn)));
      m = mn;
    }
  }

  // remainder (and unaligned fallback): plain guarded loads
  for (int i = T * TILE_FLOATS + tid; i < n; i += BLOCK) {
    float y = src[i] * K_S_LOG2E;
    lse_combine(m, s, y, 1.0f);
  }

  // wave32 shuffle reduction
  for (int off = 16; off > 0; off >>= 1) {
    float om = __shfl_xor(m, off, 32);
    float os = __shfl_xor(s, off, 32);
    lse_combine(m, s, om, os);
  }

  const int wave = tid >> 5;
  if ((tid & 31) == 0) { red_m[wave] = m; red_s[wave] = s; }
  __syncthreads();

  if (tid == 0) {
    float M0 = red_m[0], S0 = red_s[0];
#pragma unroll
    for (int w = 1; w < BLOCK / 32; ++w) lse_combine(M0, S0, red_m[w], red_s[w]);
    partial[(size_t)row * CHUNKS + chunk] = make_float2(M0, S0);
  }
}

// ---------------------------------------------------------------------------
// Phase 2: combine chunk partials, apply ArcFace margin fix on the target
// term, reduce to the mean loss.
// ---------------------------------------------------------------------------
__global__ __launch_bounds__(512)
void arc_loss_final_kernel(const float* __restrict__ cosT,
                           const int* __restrict__ target,
                           const float2* __restrict__ partial,
                           float* __restrict__ out,
                           int B, int C)
{
  __shared__ float red[512];
  float acc = 0.0f;

  for (int b = threadIdx.x; b < B; b += blockDim.x) {
    float m = -3.0e38f, s = 0.0f;
#pragma unroll
    for (int c = 0; c < CHUNKS; ++c) {
      float2 p = partial[(size_t)b * CHUNKS + c];
      lse_combine(m, s, p.x, p.y);
    }
    const int t   = target[b];
    const float ct = cosT[(size_t)b * (size_t)C + t];
    float st  = sqrtf(fmaxf(0.0f, fminf(1.0f, 1.0f - ct * ct)));
    float phi = ct * ARC_COS_M - st * ARC_SIN_M;
    phi = (ct > ARC_TH) ? phi : (ct - ARC_MM);

    const float y_cos = ct  * K_S_LOG2E;   // term phase 1 accumulated
    const float y_phi = phi * K_S_LOG2E;   // term it should have been
    s = s - exp2f(y_cos - m) + exp2f(y_phi - m);   // y_cos <= m, safe
    s = fmaxf(s, 1e-37f);

    const float lse = (log2f(s) + m) * LN2F;       // logsumexp in nats
    acc += lse - ARC_S * phi;                      // -log_softmax at target
  }

  red[threadIdx.x] = acc;
  __syncthreads();
#pragma unroll
  for (int off = 256; off > 0; off >>= 1) {
    if ((int)threadIdx.x < off) red[threadIdx.x] += red[threadIdx.x + off];
    __syncthreads();
  }
  if (threadIdx.x == 0) out[0] = red[0] / (float)B;
}

// ---------------------------------------------------------------------------
extern "C" void kernel_launch(void* const* d_in, const int* in_sizes, int n_in,
                              void* d_out, int out_size, void* d_ws, size_t ws_size,
                              hipStream_t stream) {
  const float* cosT   = (const float*)d_in[0];
  const int*   target = (const int*)d_in[1];
  float*       out    = (float*)d_out;
  float2*      part   = (float2*)d_ws;   // B * CHUNKS float2 = 32 KB

  const int total = in_sizes[0];
  const int B     = in_sizes[1];
  const int C     = total / B;

  // chunk length rounded up to a multiple of 4 floats (16B) for async b128
  const int chunkLen = (((C + CHUNKS - 1) / CHUNKS) + 3) & ~3;

  dim3 grid(CHUNKS, B);
  arc_lse_partial_kernel<<<grid, BLOCK, 0, stream>>>(cosT, part, C, chunkLen);
  arc_loss_final_kernel<<<1, 512, 0, stream>>>(cosT, target, part, out, B, C);
}